// tsgcn_attention_block_80917183857420
// MI455X (gfx1250) — compile-verified
//
#include <hip/hip_runtime.h>
#include <float.h>
#include <math.h>

#define NPTS 8192
#define DCH  32
#define KNN  24
#define BB   2
#define MTOT (BB * KNN * NPTS)   // 393216 BN population count

typedef float v2f __attribute__((ext_vector_type(2)));
typedef float v8f __attribute__((ext_vector_type(8)));

// ---------------------------------------------------------------------------
// Kernel 1: build reinterpreted Xr[b][i][j] = xc[b][j%32][i*256 + j/32]
// (torch .view quirk) and column norms xx[b][j] = sum_i Xr^2.
// ---------------------------------------------------------------------------
__global__ void k_prep(const float* __restrict__ xc,
                       float* __restrict__ Xr, float* __restrict__ xx) {
  int tid = blockIdx.x * blockDim.x + threadIdx.x;
  if (tid >= BB * NPTS) return;
  int b = tid / NPTS, j = tid % NPTS;
  int c0 = j & 31, j32 = j >> 5;
  const float* src = xc + (size_t)b * DCH * NPTS + (size_t)c0 * NPTS + j32;
  float* dst = Xr + (size_t)b * DCH * NPTS + j;
  float ss = 0.f;
#pragma unroll
  for (int i = 0; i < DCH; i++) {
    float v = src[i * 256];
    dst[(size_t)i * NPTS] = v;
    ss += v * v;
  }
  xx[tid] = ss;
}

// ---------------------------------------------------------------------------
// Kernel 2: kNN. One wave (32 threads) per 16-row tile. Gram via fp32 WMMA
// (16x16 output tile, K=32 as 8 chained V_WMMA_F32_16X16X4_F32), then
// pd = 2G - xx_n - xx_m, then sorted top-24 insertion per row in LDS.
// Tie handling matches jax.lax.top_k (earlier index wins on equal value).
// ---------------------------------------------------------------------------
__global__ void __launch_bounds__(32) k_knn(const float* __restrict__ Xr,
                                            const float* __restrict__ xx,
                                            int* __restrict__ kidx) {
  __shared__ float ptile[16 * 17];
  __shared__ float xxn_s[16];
  __shared__ float tv[16 * KNN];
  __shared__ int   ti[16 * KNN];

  int b    = blockIdx.x >> 9;          // 512 tiles per batch
  int n0   = (blockIdx.x & 511) << 4;
  int lane = threadIdx.x;
  int half = lane >> 4, lc = lane & 15;

  const float* Xbase = Xr + (size_t)b * DCH * NPTS;

  // A fragments for this n-tile: A[lane][v] = Xr[d = 4kc + v + 2*half][n0+lc]
  const float* Xa = Xbase + n0 + lc;
  v2f af[8];
#pragma unroll
  for (int kc = 0; kc < 8; kc++) {
    int d0 = 4 * kc + 2 * half;
    af[kc].x = Xa[(size_t)d0 * NPTS];
    af[kc].y = Xa[(size_t)(d0 + 1) * NPTS];
  }

  if (lane < 16) {
    xxn_s[lane] = xx[b * NPTS + n0 + lane];
    for (int t = 0; t < KNN; t++) { tv[lane * KNN + t] = -FLT_MAX; ti[lane * KNN + t] = 0; }
  }
  __syncthreads();

  float minv = -FLT_MAX;
  for (int m0 = 0; m0 < NPTS; m0 += 16) {
    const float* Xb = Xbase + m0 + lc;
    v8f acc = {0.f, 0.f, 0.f, 0.f, 0.f, 0.f, 0.f, 0.f};
#pragma unroll
    for (int kc = 0; kc < 8; kc++) {
      int d0 = 4 * kc + 2 * half;
      v2f bf;
      bf.x = Xb[(size_t)d0 * NPTS];
      bf.y = Xb[(size_t)(d0 + 1) * NPTS];
      acc = __builtin_amdgcn_wmma_f32_16x16x4_f32(false, af[kc], false, bf,
                                                  (short)0, acc, false, false);
    }
    float xxm = xx[b * NPTS + m0 + lc];
#pragma unroll
    for (int v = 0; v < 8; v++) {
      int row = v + 8 * half;
      ptile[row * 17 + lc] = 2.0f * acc[v] - xxn_s[row] - xxm;
    }
    __syncthreads();
    if (lane < 16) {
      for (int jc = 0; jc < 16; jc++) {
        float val = ptile[lane * 17 + jc];
        if (val > minv) {
          int p = KNN - 1;
          while (p > 0 && tv[lane * KNN + p - 1] < val) {
            tv[lane * KNN + p] = tv[lane * KNN + p - 1];
            ti[lane * KNN + p] = ti[lane * KNN + p - 1];
            p--;
          }
          tv[lane * KNN + p] = val;
          ti[lane * KNN + p] = m0 + jc;
          minv = tv[lane * KNN + KNN - 1];
        }
      }
    }
    __syncthreads();
  }

  if (lane < 16) {
    int base = (b * NPTS + n0 + lane) * KNN;
    for (int t = 0; t < KNN; t++) kidx[base + t] = ti[lane * KNN + t];
  }
}

// ---------------------------------------------------------------------------
// Kernel 3: BN stats pass. One wave per point n; lane = out-channel.
// y-column j = jq*N + n: tile half uses point n directly, neighbor half uses
// g = idx_flat[b][j]. Accumulate per-channel sum/sumsq (incl. conv bias) as
// deterministic per-block partials.
// ---------------------------------------------------------------------------
__global__ void __launch_bounds__(64) k_stats(const float* __restrict__ xc,
                                              const float* __restrict__ xn,
                                              const float* __restrict__ w0, const float* __restrict__ b0,
                                              const float* __restrict__ w1, const float* __restrict__ b1,
                                              const float* __restrict__ w2, const float* __restrict__ b2,
                                              const int* __restrict__ kidx,
                                              float* __restrict__ gpart) {
  __shared__ float wT[3 * 64 * 32];   // transposed weights: wT[cv][c][o]
  __shared__ float cb[3 * 32];
  __shared__ float tile_xc[2][32], tile_xn[2][32], nb_xc[2][32], nb_xn[2][32];
  __shared__ float red[2][6][32];

  int tx = threadIdx.x;
  for (int t = tx; t < 3 * 2048; t += 64) {
    int cv = t >> 11, r = t & 2047, c = r >> 5, o = r & 31;
    const float* W = (cv == 0) ? w0 : ((cv == 1) ? w1 : w2);
    wT[t] = W[o * 64 + c];
  }
  for (int t = tx; t < 96; t += 64) {
    int cv = t >> 5, o = t & 31;
    const float* Bv = (cv == 0) ? b0 : ((cv == 1) ? b1 : b2);
    cb[t] = Bv[o];
  }

  int w = tx >> 5, o = tx & 31;
  int gw = blockIdx.x * 2 + w;
  int b = gw / NPTS, n = gw % NPTS;
  const float* xcb = xc + (size_t)b * DCH * NPTS + (size_t)o * NPTS;
  const float* xnb = xn + (size_t)b * DCH * NPTS + (size_t)o * NPTS;
  tile_xc[w][o] = xcb[n];
  tile_xn[w][o] = xnb[n];
  __syncthreads();

  const float* wT0 = wT;
  const float* wT1 = wT + 2048;
  const float* wT2 = wT + 4096;

  float s0 = 0, q0 = 0, s1 = 0, q1 = 0, s2 = 0, q2 = 0;
  for (int jq = 0; jq < KNN; jq++) {
    int g = kidx[(size_t)b * NPTS * KNN + (size_t)jq * NPTS + n];
    nb_xc[w][o] = xcb[g];
    nb_xn[w][o] = xnb[g];
    __syncthreads();
    float l0 = 0, l1 = 0, l2 = 0;
#pragma unroll 8
    for (int c = 0; c < 32; c++) {
      float tc = tile_xc[w][c], tn = tile_xn[w][c];
      float nc = nb_xc[w][c],  nn = nb_xn[w][c];
      l0 += wT0[c * 32 + o] * tc + wT0[(32 + c) * 32 + o] * nc;
      l1 += wT1[c * 32 + o] * tn + wT1[(32 + c) * 32 + o] * (tn - nn);
      l2 += wT2[c * 32 + o] * tn + wT2[(32 + c) * 32 + o] * nn;
    }
    float y0 = l0 + cb[o], y1 = l1 + cb[32 + o], y2 = l2 + cb[64 + o];
    s0 += y0; q0 += y0 * y0;
    s1 += y1; q1 += y1 * y1;
    s2 += y2; q2 += y2 * y2;
    __syncthreads();
  }
  red[w][0][o] = s0; red[w][1][o] = q0;
  red[w][2][o] = s1; red[w][3][o] = q1;
  red[w][4][o] = s2; red[w][5][o] = q2;
  __syncthreads();
  for (int t = tx; t < 192; t += 64) {
    int idx6 = t / 32, oo = t % 32;
    gpart[(size_t)blockIdx.x * 192 + t] = red[0][idx6][oo] + red[1][idx6][oo];
  }
}

// ---------------------------------------------------------------------------
// Kernel 4: fold partials -> BN mean/var -> per-channel affine (bias folded):
// out = lrelu(lin*scale + shift), scale = g*rsqrt(var+eps),
// shift = (conv_bias - mean)*scale + beta.
// ---------------------------------------------------------------------------
__global__ void __launch_bounds__(256) k_reduce(const float* __restrict__ gpart,
    const float* __restrict__ b0, const float* __restrict__ g0, const float* __restrict__ be0,
    const float* __restrict__ b1, const float* __restrict__ g1, const float* __restrict__ be1,
    const float* __restrict__ b2, const float* __restrict__ g2, const float* __restrict__ be2,
    float* __restrict__ sparams, int nblocks) {
  __shared__ float tot[192];
  int t = threadIdx.x;
  if (t < 192) {
    float s = 0.f;
    for (int blk = 0; blk < nblocks; blk++) s += gpart[(size_t)blk * 192 + t];
    tot[t] = s;
  }
  __syncthreads();
  if (t < 96) {
    int cv = t / 32, o = t % 32;
    float S = tot[cv * 64 + o];
    float Q = tot[cv * 64 + 32 + o];
    float m   = S / (float)MTOT;
    float var = Q / (float)MTOT - m * m;
    float inv = rsqrtf(var + 1e-5f);
    const float* G  = (cv == 0) ? g0  : ((cv == 1) ? g1  : g2);
    const float* Be = (cv == 0) ? be0 : ((cv == 1) ? be1 : be2);
    const float* Cb = (cv == 0) ? b0  : ((cv == 1) ? b1  : b2);
    float sc = G[o] * inv;
    sparams[t]      = sc;
    sparams[96 + t] = (Cb[o] - m) * sc + Be[o];
  }
}

// ---------------------------------------------------------------------------
// Kernel 5: fused pass 2. One wave per point n. For jq=0..23 compute the three
// conv columns at j = jq*N + n, apply affine+LeakyReLU, and remap through the
// (o,jq) -> v = o*24+jq -> (kk=v/32, o2=v%32) reshape view. Then per kk:
// stable softmax over o2 (wave shuffles) for attention-weighted sum (xc1),
// and max over kk for branch 2 (xn1).
// ---------------------------------------------------------------------------
__global__ void __launch_bounds__(64) k_fuse(const float* __restrict__ xc,
                                             const float* __restrict__ xn,
                                             const float* __restrict__ w0,
                                             const float* __restrict__ w1,
                                             const float* __restrict__ w2,
                                             const int* __restrict__ kidx,
                                             const float* __restrict__ sparams,
                                             float* __restrict__ out) {
  __shared__ float wT[3 * 64 * 32];
  __shared__ float tile_xc[2][32], tile_xn[2][32], nb_xc[2][32], nb_xn[2][32];
  __shared__ float Abuf[2][KNN * 32], Fbuf[2][KNN * 32], Mbuf[2][KNN * 32];

  int tx = threadIdx.x;
  for (int t = tx; t < 3 * 2048; t += 64) {
    int cv = t >> 11, r = t & 2047, c = r >> 5, o = r & 31;
    const float* W = (cv == 0) ? w0 : ((cv == 1) ? w1 : w2);
    wT[t] = W[o * 64 + c];
  }

  int w = tx >> 5, o = tx & 31;
  int gw = blockIdx.x * 2 + w;
  int b = gw / NPTS, n = gw % NPTS;
  const float* xcb = xc + (size_t)b * DCH * NPTS + (size_t)o * NPTS;
  const float* xnb = xn + (size_t)b * DCH * NPTS + (size_t)o * NPTS;
  tile_xc[w][o] = xcb[n];
  tile_xn[w][o] = xnb[n];

  float sc0 = sparams[o],      sh0 = sparams[96 + o];
  float sc1 = sparams[32 + o], sh1 = sparams[96 + 32 + o];
  float sc2 = sparams[64 + o], sh2 = sparams[96 + 64 + o];
  __syncthreads();

  const float* wT0 = wT;
  const float* wT1 = wT + 2048;
  const float* wT2 = wT + 4096;

  for (int jq = 0; jq < KNN; jq++) {
    int g = kidx[(size_t)b * NPTS * KNN + (size_t)jq * NPTS + n];
    nb_xc[w][o] = xcb[g];
    nb_xn[w][o] = xnb[g];
    __syncthreads();
    float l0 = 0, l1 = 0, l2 = 0;
#pragma unroll 8
    for (int c = 0; c < 32; c++) {
      float tc = tile_xc[w][c], tn = tile_xn[w][c];
      float nc = nb_xc[w][c],  nn = nb_xn[w][c];
      l0 += wT0[c * 32 + o] * tc + wT0[(32 + c) * 32 + o] * nc;
      l1 += wT1[c * 32 + o] * tn + wT1[(32 + c) * 32 + o] * (tn - nn);
      l2 += wT2[c * 32 + o] * tn + wT2[(32 + c) * 32 + o] * nn;
    }
    float a0 = l0 * sc0 + sh0; a0 = (a0 > 0.f) ? a0 : 0.01f * a0;  // f01
    float a1 = l1 * sc1 + sh1; a1 = (a1 > 0.f) ? a1 : 0.01f * a1;  // attn
    float a2 = l2 * sc2 + sh2; a2 = (a2 > 0.f) ? a2 : 0.01f * a2;  // f11
    int v = o * KNN + jq;   // flat index in the (k,32) view: v = kk*32 + o2
    Fbuf[w][v] = a0;
    Abuf[w][v] = a1;
    Mbuf[w][v] = a2;
    __syncthreads();
  }

  // Branch 2: max over kk (lane = o2)
  float mx = -FLT_MAX;
  for (int kk = 0; kk < KNN; kk++) mx = fmaxf(mx, Mbuf[w][kk * 32 + o]);
  out[(size_t)BB * DCH * NPTS + (size_t)b * DCH * NPTS + (size_t)o * NPTS + n] = mx;

  // Branch 0: softmax over o2 per kk, attention-weighted sum over kk
  float accf = 0.f;
  for (int kk = 0; kk < KNN; kk++) {
    float av = Abuf[w][kk * 32 + o];
    float m_ = av;
    for (int off = 16; off > 0; off >>= 1) m_ = fmaxf(m_, __shfl_xor(m_, off, 32));
    float e = expf(av - m_);
    float se = e;
    for (int off = 16; off > 0; off >>= 1) se += __shfl_xor(se, off, 32);
    accf += (e / se) * Fbuf[w][kk * 32 + o];
  }
  out[(size_t)b * DCH * NPTS + (size_t)o * NPTS + n] = accf;
}

// ---------------------------------------------------------------------------
extern "C" void kernel_launch(void* const* d_in, const int* in_sizes, int n_in,
                              void* d_out, int out_size, void* d_ws, size_t ws_size,
                              hipStream_t stream) {
  (void)in_sizes; (void)n_in; (void)out_size; (void)ws_size;
  const float* xc  = (const float*)d_in[0];
  const float* xn  = (const float*)d_in[1];
  const float* w0  = (const float*)d_in[2];
  const float* b0  = (const float*)d_in[3];
  const float* g0  = (const float*)d_in[4];
  const float* be0 = (const float*)d_in[5];
  const float* w1  = (const float*)d_in[6];
  const float* b1  = (const float*)d_in[7];
  const float* g1  = (const float*)d_in[8];
  const float* be1 = (const float*)d_in[9];
  const float* w2  = (const float*)d_in[10];
  const float* b2  = (const float*)d_in[11];
  const float* g2  = (const float*)d_in[12];
  const float* be2 = (const float*)d_in[13];

  // workspace layout (~10 MB total)
  float* Xr      = (float*)d_ws;                         // B*32*N
  float* xx      = Xr + (size_t)BB * DCH * NPTS;         // B*N
  int*   kidx    = (int*)(xx + (size_t)BB * NPTS);       // B*N*24
  float* gpart   = (float*)(kidx + (size_t)BB * NPTS * KNN);  // 8192*192
  float* sparams = gpart + (size_t)(BB * NPTS / 2) * 192;     // 192

  int nStatBlocks = BB * NPTS / 2;  // 8192 blocks, 2 waves each (one per point)

  k_prep  <<<(BB * NPTS + 255) / 256, 256, 0, stream>>>(xc, Xr, xx);
  k_knn   <<<BB * (NPTS / 16), 32, 0, stream>>>(Xr, xx, kidx);
  k_stats <<<nStatBlocks, 64, 0, stream>>>(xc, xn, w0, b0, w1, b1, w2, b2, kidx, gpart);
  k_reduce<<<1, 256, 0, stream>>>(gpart, b0, g0, be0, b1, g1, be1, b2, g2, be2,
                                  sparams, nStatBlocks);
  k_fuse  <<<nStatBlocks, 64, 0, stream>>>(xc, xn, w0, w1, w2, kidx, sparams,
                                           (float*)d_out);
}